// SRenderY_57518202028323
// MI455X (gfx1250) — compile-verified
//
#include <hip/hip_runtime.h>

// Problem constants (match the reference)
#define PN 8
#define PH 512
#define PW 512
#define PD 3
#define PHW (PH * PW)
#define NPIX (PN * PHW)

// Barycentric interpolation of per-face vertex attributes (PyTorch3D-style).
// Pure gather / bandwidth-bound kernel:
//   - pix_to_face (int64) and bary (f32x3) streamed once  -> non-temporal loads
//   - attributes table (28.8 MB, fits in 192MB L2)        -> regular-temporal gather + prefetch
//   - output planes streamed once                          -> non-temporal stores
__global__ __launch_bounds__(256) void SRenderY_interp_kernel(
    const long long* __restrict__ pix_to_face,  // [N,H,W,1] int64
    const float*     __restrict__ bary,         // [N,H,W,1,3] f32
    const float*     __restrict__ attrs,        // [N*F, 3, 3] f32 (packed)
    float*           __restrict__ out)          // [N, 4, H, W] f32
{
    const int gid = blockIdx.x * blockDim.x + threadIdx.x;
    if (gid >= NPIX) return;

    const int n  = gid / PHW;
    const int hw = gid - n * PHW;

    // Streamed once: non-temporal so it doesn't evict the L2-resident attr table.
    const long long idx = __builtin_nontemporal_load(&pix_to_face[gid]);
    const bool valid = (idx >= 0);
    const long long safe = valid ? idx : 0;  // reference does a safe gather at 0

    // Kick the gather cacheline early; attr table lives in L2 (RT temporal).
    const float* __restrict__ a = attrs + (size_t)safe * 9;
    __builtin_prefetch(a, 0, 3);  // -> global_prefetch_b8

    // Bary coords: streamed once, non-temporal.
    const float* __restrict__ b = bary + (size_t)gid * 3;
    const float b0 = __builtin_nontemporal_load(&b[0]);
    const float b1 = __builtin_nontemporal_load(&b[1]);
    const float b2 = __builtin_nontemporal_load(&b[2]);

    // Gather 3 verts x 3 attrs (regular-temporal: reused across pixels via L2).
    const float a00 = a[0], a01 = a[1], a02 = a[2];
    const float a10 = a[3], a11 = a[4], a12 = a[5];
    const float a20 = a[6], a21 = a[7], a22 = a[8];

    float r0 = fmaf(b0, a00, fmaf(b1, a10, b2 * a20));
    float r1 = fmaf(b0, a01, fmaf(b1, a11, b2 * a21));
    float r2 = fmaf(b0, a02, fmaf(b1, a12, b2 * a22));

    // Branchless mask (empty pixels are ~1e-6 of all pixels).
    const float vis = valid ? 1.0f : 0.0f;
    r0 = valid ? r0 : 0.0f;
    r1 = valid ? r1 : 0.0f;
    r2 = valid ? r2 : 0.0f;

    // Output planes [N,4,H,W]: write-once -> non-temporal stores.
    float* __restrict__ o = out + ((size_t)n * 4) * PHW + hw;
    __builtin_nontemporal_store(r0,  &o[0 * PHW]);
    __builtin_nontemporal_store(r1,  &o[1 * PHW]);
    __builtin_nontemporal_store(r2,  &o[2 * PHW]);
    __builtin_nontemporal_store(vis, &o[3 * PHW]);
}

extern "C" void kernel_launch(void* const* d_in, const int* in_sizes, int n_in,
                              void* d_out, int out_size, void* d_ws, size_t ws_size,
                              hipStream_t stream) {
    (void)in_sizes; (void)n_in; (void)out_size; (void)d_ws; (void)ws_size;

    const long long* pix_to_face = (const long long*)d_in[0];  // int64 per reference
    const float*     bary        = (const float*)d_in[1];
    const float*     attrs       = (const float*)d_in[2];
    float*           out         = (float*)d_out;

    const int threads = 256;                       // 8 wave32 waves per block
    const int blocks  = (NPIX + threads - 1) / threads;
    SRenderY_interp_kernel<<<blocks, threads, 0, stream>>>(pix_to_face, bary, attrs, out);
}